// NegativeLogLikelihood_33638183862568
// MI455X (gfx1250) — compile-verified
//
#include <hip/hip_runtime.h>
#include <hip/hip_bf16.h>
#include <math.h>

// ---------------------------------------------------------------------------
// Cox proportional-hazards negative log-likelihood + L2(W) for MI455X/gfx1250.
//
// Heavy part: num[j] = sum_i exp(r[i]) * (y[i] >= y[j]); den[j] = |risk set|.
// Mapped onto V_WMMA_F32_16X16X4_F32:  D(16x16) = A(16x4) x B(4x16) + C
//   A row0 = exp(r[k]), row1 = 1.0, rows 2..15 = 0      (k = K dimension)
//   B[k,n] = (y[k] >= y[jbase+n]) ? 1 : 0               (mask tile)
//   => D[0,n] accumulates num, D[1,n] accumulates den, in full f32.
// One wave = one 16-wide j tile, sweeping all k through LDS-staged chunks.
// ---------------------------------------------------------------------------

typedef __attribute__((ext_vector_type(2))) float v2f;
typedef __attribute__((ext_vector_type(8))) float v8f;

#define KC    2048          // k-chunk staged in LDS (8 KB y + 8 KB exp)
#define WAVES 4             // wave32 waves per block; one j-tile per wave
#define TPB   (WAVES * 32)

__global__ __launch_bounds__(TPB) void cox_wmma_kernel(
    const float* __restrict__ r,      // risk_pred (N)
    const float* __restrict__ y,      // survival times (N)
    const int*   __restrict__ e,      // event indicator (N)
    float*       __restrict__ term,   // out: per-sample e*(r - log(num/den)) (N)
    int n)
{
  __shared__ float sy[KC];    // y chunk
  __shared__ float sex[KC];   // exp(r) chunk

  const int tid  = threadIdx.x;
  const int lane = tid & 31;          // wave32
  const int wv   = tid >> 5;
  const int jbase  = (blockIdx.x * WAVES + wv) * 16;
  const bool jvalid = (jbase < n);

  // Per-lane column y value: B/D column n = lane&15  ->  y[jbase + (lane&15)]
  int jj = jbase + (lane & 15);
  if (jj >= n) jj = n - 1;
  const float yj = jvalid ? y[jj] : 0.0f;

  // WMMA A (16x4 f32) layout: lane holds row M=lane&15; lanes<16 cover K={0,1},
  // lanes>=16 cover K={2,3}. Same K-pair selection applies to B rows.
  const int  half  = (lane < 16) ? 0 : 2;   // K offset for this lane half
  const int  row   = lane & 15;
  const bool isRow0 = (row == 0);
  const float aConst = (row == 1) ? 1.0f : 0.0f;  // ones row / zero rows

  v8f acc = {0.f, 0.f, 0.f, 0.f, 0.f, 0.f, 0.f, 0.f};

  const int nChunks = (n + KC - 1) / KC;
  for (int c = 0; c < nChunks; ++c) {
    const int base = c * KC;
    // Stage this k-chunk: y and exp(r). Pad with y=-inf (mask=0), exp=0.
    for (int idx = tid; idx < KC; idx += TPB) {
      const int g = base + idx;
      if (g < n) { sy[idx] = y[g];     sex[idx] = expf(r[g]); }
      else       { sy[idx] = -INFINITY; sex[idx] = 0.0f; }
    }
    __syncthreads();

    if (jvalid) {   // wave-uniform branch: EXEC all-ones inside (WMMA req.)
      const float* ap = &sex[half];
      const float* bp = &sy[half];
      #pragma unroll 8
      for (int kb = 0; kb < KC; kb += 4) {
        // A operand: broadcast ds_load_b64 of exp pair, selected per row.
        const v2f ev = *(const v2f*)ap;
        v2f a;
        a.x = isRow0 ? ev.x : aConst;
        a.y = isRow0 ? ev.y : aConst;
        // B operand: mask(k, j) = (y[k] >= y[j]) as 0.0/1.0.
        const v2f yk = *(const v2f*)bp;
        v2f b;
        b.x = (yk.x >= yj) ? 1.0f : 0.0f;
        b.y = (yk.y >= yj) ? 1.0f : 0.0f;
        // D = A*B + C, f32 accumulate. (neg_a, A, neg_b, B, c_mod, C, reuse_a, reuse_b)
        acc = __builtin_amdgcn_wmma_f32_16x16x4_f32(
            false, a, false, b, (short)0, acc, false, false);
        ap += 4;
        bp += 4;
      }
    }
    __syncthreads();
  }

  // D layout: VGPR0 lanes 0-15 = M=0 (num), VGPR1 lanes 0-15 = M=1 (den).
  if (jvalid && lane < 16 && (jbase + lane) < n) {
    const int j = jbase + lane;
    const float num = acc[0];
    const float den = acc[1];
    const float ef  = (float)e[j];
    term[j] = (ef != 0.0f) ? (r[j] - logf(num / den)) : 0.0f;
  }
}

// ---------------------------------------------------------------------------
// Finalize: loss = -(sum_j term[j]) / (sum_j e[j]) + 0.01 * sqrt(sum W^2).
// Single block, fixed-order reductions -> deterministic.
// ---------------------------------------------------------------------------
__global__ __launch_bounds__(256) void cox_finalize_kernel(
    const float* __restrict__ term, const int* __restrict__ e,
    const float* __restrict__ W, float* __restrict__ out, int n, int nw)
{
  __shared__ float sT[256];
  __shared__ float sE[256];
  __shared__ float sW[256];
  const int tid = threadIdx.x;
  float aT = 0.f, aE = 0.f, aW = 0.f;
  for (int i = tid; i < n; i += 256) {
    aT += term[i];
    aE += (float)e[i];
  }
  for (int i = tid; i < nw; i += 256) {
    const float w = W[i];
    aW += w * w;
  }
  sT[tid] = aT; sE[tid] = aE; sW[tid] = aW;
  __syncthreads();
  for (int s = 128; s > 0; s >>= 1) {
    if (tid < s) {
      sT[tid] += sT[tid + s];
      sE[tid] += sE[tid + s];
      sW[tid] += sW[tid + s];
    }
    __syncthreads();
  }
  if (tid == 0) {
    out[0] = -(sT[0] / sE[0]) + 0.01f * sqrtf(sW[0]);
  }
}

// ---------------------------------------------------------------------------
extern "C" void kernel_launch(void* const* d_in, const int* in_sizes, int n_in,
                              void* d_out, int out_size, void* d_ws, size_t ws_size,
                              hipStream_t stream) {
  const float* risk = (const float*)d_in[0];   // risk_pred (N,1) f32
  const float* y    = (const float*)d_in[1];   // y (N,1) f32
  const int*   e    = (const int*)d_in[2];     // e (N,1) i32
  const float* W    = (const float*)d_in[3];   // W (512,256) f32
  float* out = (float*)d_out;                  // scalar f32

  const int n  = in_sizes[0];                  // 16384
  const int nw = in_sizes[3];                  // 131072

  float* term = (float*)d_ws;                  // n floats of scratch (64 KB)

  const int jtiles = (n + 15) / 16;
  const int blocks = (jtiles + WAVES - 1) / WAVES;

  cox_wmma_kernel<<<blocks, TPB, 0, stream>>>(risk, y, e, term, n);
  cox_finalize_kernel<<<1, 256, 0, stream>>>(term, e, W, out, n, nw);
}